// PatchEmbedder_69269232550118
// MI455X (gfx1250) — compile-verified
//
#include <hip/hip_runtime.h>
#include <stdint.h>

typedef uint32_t u32;
typedef __attribute__((ext_vector_type(4))) uint32_t u32x4;
typedef __attribute__((ext_vector_type(8))) uint32_t u32x8;
typedef __attribute__((ext_vector_type(4))) float    f32x4;

#define WAVES_PER_BLOCK 4
#define LL 16            // ids per patch per table
#define DD 128           // embedding dim
#define TILE_FLOATS (LL * DD)        // 2048 floats = 8KB per gathered tile
#define NPATCH (16 * 2048)           // B * P

// Accumulate one gathered 16x128 tile from LDS into acc (lane owns 4 floats).
__device__ __forceinline__ void accum_tile(const float* buf, u32 lane, f32x4& acc) {
    const f32x4* src = (const f32x4*)buf + lane;   // float4 index: row*32 + lane
#pragma unroll
    for (int r = 0; r < LL; ++r) {
        f32x4 v = src[r * (DD / 4)];
        acc.x += v.x; acc.y += v.y; acc.z += v.z; acc.w += v.w;
    }
}

// Issue one TDM gather: 16 rows of 512B from `emb` (row indices in pk[0..7],
// packed 2x16-bit per dword) into LDS at byte offset lds_addr.
__device__ __forceinline__ void tdm_gather16(const float* emb, u32 lds_addr,
                                             const u32* pk) {
    u32x4 g0;
    uint64_t ga = (uint64_t)(uintptr_t)emb;
    g0[0] = 0x80000001u;                                   // gather_mode=1, 16-bit idx, count=1
    g0[1] = lds_addr;                                      // LDS dest (bytes)
    g0[2] = (u32)(ga & 0xffffffffu);                       // global_addr[31:0]
    g0[3] = (u32)((ga >> 32) & 0x01ffffffu) | 0x80000000u; // addr[56:32] | type=2

    u32x8 g1;
    g1[0] = 0x00020000u;          // workgroup_mask=0, data_size=2 (4 bytes)
    g1[1] = (u32)DD << 16;        // tensor_dim0 = 128 (low 16 bits of field)
    g1[2] = 0u;                   // tensor_dim0 hi / tensor_dim1 lo16 (65536 -> 0)
    g1[3] = ((u32)DD << 16) | 1u; // tile_dim0 = 128 | tensor_dim1 hi16 (=1 -> 65536)
    g1[4] = (u32)LL;              // tile_dim1 = 16 valid gather indices
    g1[5] = (u32)DD;              // tensor_dim0_stride = 128 elements
    g1[6] = 0u;
    g1[7] = 0u;

    u32x4 g2 = { pk[0], pk[1], pk[2], pk[3] };  // row_index_0..7
    u32x4 g3 = { pk[4], pk[5], pk[6], pk[7] };  // row_index_8..15

    asm volatile("tensor_load_to_lds %0, %1, %2, %3"
                 :: "s"(g0), "s"(g1), "s"(g2), "s"(g3)
                 : "memory");
}

__global__ void __launch_bounds__(WAVES_PER_BLOCK * 32, 1)
patch_embed_tdm(const int* __restrict__ ids1,
                const int* __restrict__ ids2,
                const int* __restrict__ ids3,
                const unsigned char* __restrict__ mask,
                const float* __restrict__ emb1,
                const float* __restrict__ emb2,
                const float* __restrict__ emb3,
                float* __restrict__ out) {
    __shared__ float smem[WAVES_PER_BLOCK * 3 * TILE_FLOATS];  // 96 KB

    const u32 lane  = threadIdx.x & 31u;
    const u32 wave  = __builtin_amdgcn_readfirstlane(threadIdx.x >> 5);
    const u32 patch = blockIdx.x * WAVES_PER_BLOCK + wave;     // wave-uniform

    float* obase = out + (size_t)patch * DD + (size_t)lane * 4;

    // Uniform mask test: masked patches contribute zero; skip all gathers.
    if (mask[patch] == 0u) {
        f32x4 z = {0.f, 0.f, 0.f, 0.f};
        *(f32x4*)obase = z;
        return;
    }

    float* buf = &smem[wave * 3 * TILE_FLOATS];
    const u32 lds_base = (u32)(uintptr_t)buf;  // flat LDS addr -> low 32 bits = LDS byte offset

    const int*   idtab[3] = { ids1, ids2, ids3 };
    const float* etab[3]  = { emb1, emb2, emb3 };

    // Issue all three gather descriptors back-to-back (TENSORcnt = 3 in flight).
#pragma unroll
    for (int t = 0; t < 3; ++t) {
        const int* idp = idtab[t] + (size_t)patch * LL;  // uniform -> SMEM loads
        u32 pk[8];
#pragma unroll
        for (int k = 0; k < 8; ++k) {
            u32 a = (u32)idp[2 * k];
            u32 b = (u32)idp[2 * k + 1];
            pk[k] = (a & 0xffffu) | (b << 16);           // two 16-bit row indices
        }
        tdm_gather16(etab[t], lds_base + (u32)t * (TILE_FLOATS * 4), pk);
    }

    // Drain in order, overlapping accumulate of tile t with tiles t+1.. in flight.
    f32x4 acc = {0.f, 0.f, 0.f, 0.f};
    __builtin_amdgcn_s_wait_tensorcnt(2);
    accum_tile(buf + 0 * TILE_FLOATS, lane, acc);
    __builtin_amdgcn_s_wait_tensorcnt(1);
    accum_tile(buf + 1 * TILE_FLOATS, lane, acc);
    __builtin_amdgcn_s_wait_tensorcnt(0);
    accum_tile(buf + 2 * TILE_FLOATS, lane, acc);

    *(f32x4*)obase = acc;
}

extern "C" void kernel_launch(void* const* d_in, const int* in_sizes, int n_in,
                              void* d_out, int out_size, void* d_ws, size_t ws_size,
                              hipStream_t stream) {
    const int*           ids1 = (const int*)d_in[0];
    const int*           ids2 = (const int*)d_in[1];
    const int*           ids3 = (const int*)d_in[2];
    const unsigned char* msk  = (const unsigned char*)d_in[3];  // jnp.bool_ -> 1 byte
    const float*         emb1 = (const float*)d_in[4];
    const float*         emb2 = (const float*)d_in[5];
    const float*         emb3 = (const float*)d_in[6];
    float*               outp = (float*)d_out;

    dim3 grid(NPATCH / WAVES_PER_BLOCK);      // 8192 blocks, one patch per wave
    dim3 block(WAVES_PER_BLOCK * 32);         // 128 threads = 4 waves
    patch_embed_tdm<<<grid, block, 0, stream>>>(ids1, ids2, ids3, msk,
                                                emb1, emb2, emb3, outp);
}